// Dynamic_education_13494787244860
// MI455X (gfx1250) — compile-verified
//
#include <hip/hip_runtime.h>
#include <hip/hip_bf16.h>
#include <math.h>

// ---------------------------------------------------------------------------
// Problem constants (from reference)
// ---------------------------------------------------------------------------
#define S 8192
#define F 1024
#define A 256
#define SIGMA_X 4.0f
#define SIGMA_Y 1.0f
#define EPS_VALID 1e-8f
#define EPS_DIS 1e-4f

typedef __attribute__((ext_vector_type(2))) float v2f;
typedef __attribute__((ext_vector_type(8))) float v8f;

// ---------------------------------------------------------------------------
// Kernel 1: hp = tanh(batch_data @ W_h + b_h); hx/hy; pack node data.
// One wave (32 lanes) per row. 256 threads = 8 rows per block.
// ---------------------------------------------------------------------------
__launch_bounds__(256)
__global__ void k1_highlight(const float* __restrict__ bd,
                             const float* __restrict__ xywh,
                             const float* __restrict__ Wh,
                             const float* __restrict__ bh,
                             float* __restrict__ hp_out,   // [S,2]
                             float4* __restrict__ pos,     // [S] {X,Y,hx,hy}
                             float* __restrict__ val)      // [S] 1.0/0.0
{
    int row  = blockIdx.x * 8 + (threadIdx.x >> 5);
    int lane = threadIdx.x & 31;
    const float* a = bd + (size_t)row * F;

    float acc0 = 0.f, acc1 = 0.f;
#pragma unroll
    for (int it = 0; it < 8; ++it) {
        int k = it * 128 + lane * 4;
        float4 av  = *(const float4*)(a + k);
        float4 w01 = *(const float4*)(Wh + (size_t)k * 2);      // Wh[k],Wh[k+1]
        float4 w23 = *(const float4*)(Wh + (size_t)k * 2 + 4);  // Wh[k+2],Wh[k+3]
        acc0 += av.x * w01.x + av.y * w01.z + av.z * w23.x + av.w * w23.z;
        acc1 += av.x * w01.y + av.y * w01.w + av.z * w23.y + av.w * w23.w;
    }
#pragma unroll
    for (int off = 16; off > 0; off >>= 1) {
        acc0 += __shfl_xor(acc0, off, 32);
        acc1 += __shfl_xor(acc1, off, 32);
    }
    if (lane == 0) {
        float x = xywh[4 * row + 0];
        float y = xywh[4 * row + 1];
        float w = xywh[4 * row + 2];
        float h = xywh[4 * row + 3];
        float hp0 = tanhf(acc0 + bh[0]);
        float hp1 = tanhf(acc1 + bh[1]);
        float hx  = hp0 * w * SIGMA_X + x;
        float hy  = hp1 * h * SIGMA_Y + y;
        hp_out[2 * row + 0] = hp0;
        hp_out[2 * row + 1] = hp1;
        pos[row] = make_float4(x, y, hx, hy);
        val[row] = ((x + y + w + h) >= EPS_VALID) ? 1.f : 0.f;
    }
}

// ---------------------------------------------------------------------------
// Kernel 2: targets = relu(batch_data @ W_g + b_g)   [8192x1024]x[1024x256]
// 512 threads = 16 waves; block owns 16 rows x 256 cols; wave owns a 16x16
// N-tile. A tile staged in LDS with stride 132 (bank-conflict-free b64
// fragment reads), V_WMMA_F32_16X16X4_F32 accumulate, fused bias+ReLU.
// ---------------------------------------------------------------------------
__launch_bounds__(512)
__global__ void k2_gcn(const float* __restrict__ bd,
                       const float* __restrict__ Wg,     // [F,A] row-major
                       const float* __restrict__ bg,     // [A]
                       float* __restrict__ targets)      // [S,A]
{
    constexpr int ASTR = 132;            // padded stride: (4L+kk) sweeps banks
    __shared__ float As[16 * ASTR];      // ~8.3 KB

    int tid   = threadIdx.x;
    int wave  = tid >> 5;                // 0..15
    int lane  = tid & 31;
    int lhalf = lane >> 4;               // 0/1
    int l15   = lane & 15;
    int m0    = blockIdx.x * 16;
    int n0    = wave * 16;

    v8f c = {};
    for (int k0 = 0; k0 < F; k0 += 128) {
        __syncthreads();
        {   // stage A tile: 512 threads x float4 = 16 rows x 128 k
            int r  = tid >> 5;
            int kk = (tid & 31) * 4;
            float4 av = *(const float4*)(bd + (size_t)(m0 + r) * F + k0 + kk);
            *(float4*)(&As[r * ASTR + kk]) = av;
        }
        __syncthreads();
#pragma unroll
        for (int kc = 0; kc < 32; ++kc) {
            int kk = kc * 4;
            // A fragment: lane L -> A[L&15][kk + {0,1} + 2*(L>>4)]
            float2 af = *(const float2*)(&As[l15 * ASTR + kk + 2 * lhalf]);
            v2f a; a[0] = af.x; a[1] = af.y;
            // B fragment: lane L -> B[kk + {0,1} + 2*(L>>4)][L&15]
            const float* bp = Wg + (size_t)(k0 + kk + 2 * lhalf) * A + n0 + l15;
            v2f b; b[0] = bp[0]; b[1] = bp[A];
            c = __builtin_amdgcn_wmma_f32_16x16x4_f32(false, a, false, b,
                                                      (short)0, c, false, false);
        }
    }
#pragma unroll
    for (int r = 0; r < 8; ++r) {
        int m = m0 + r + 8 * lhalf;
        int n = n0 + l15;
        float v = c[r] + bg[n];
        targets[(size_t)m * A + n] = v > 0.f ? v : 0.f;
    }
}

// ---------------------------------------------------------------------------
// Kernel 3: fused pairwise-weight x targets GEMM with on-the-fly row sums.
// Block = 16 waves = 32 rows x 256 cols of output (wave w owns N-tile 16w
// and TWO row tiles: rows m0..15 and m0+16..31 -> one B fragment feeds two
// WMMAs). Double-buffered 32x32 w-tile in LDS (stride 36, conflict-free),
// single barrier per 32-K step; row sums accumulated in registers.
// ---------------------------------------------------------------------------
__launch_bounds__(512)
__global__ void k3_attn(const float4* __restrict__ pos,   // {X,Y,hx,hy}
                        const float* __restrict__ val,
                        const float* __restrict__ targets,
                        float* __restrict__ out)          // [S,A]
{
    constexpr int WSTR = 36;             // 36L mod 64 sweeps the banks
    __shared__ float Ws[2][32 * WSTR];   // ~9.2 KB
    __shared__ float rs_lds[32];

    int tid   = threadIdx.x;
    int wave  = tid >> 5;                // 0..15
    int lane  = tid & 31;
    int lhalf = lane >> 4;
    int l15   = lane & 15;
    int m0    = blockIdx.x * 32;
    int n0    = wave * 16;

    int i0 = m0 + wave;                  // wave generates w rows i0 and i1
    int i1 = i0 + 16;
    float4 p0 = pos[i0];
    float4 p1 = pos[i1];
    float hx0 = p0.z, hy0 = p0.w, vi0 = val[i0];
    float hx1 = p1.z, hy1 = p1.w, vi1 = val[i1];

    v8f c0 = {}, c1 = {};
    float rs0 = 0.f, rs1 = 0.f;

    // prologue: generate tile 0
    {
        float4 pj = pos[lane];
        float vj  = val[lane];
        float dx0 = hx0 - pj.x, dy0 = hy0 - pj.y;
        float dx1 = hx1 - pj.x, dy1 = hy1 - pj.y;
        float w0 = (vi0 * vj) / (sqrtf(dx0 * dx0 + dy0 * dy0) + EPS_DIS);
        float w1 = (vi1 * vj) / (sqrtf(dx1 * dx1 + dy1 * dy1) + EPS_DIS);
        Ws[0][wave * WSTR + lane]        = w0;
        Ws[0][(wave + 16) * WSTR + lane] = w1;
        rs0 += w0; rs1 += w1;
    }
    __syncthreads();

    constexpr int NIT = S / 32;          // 256 K-steps
    for (int t = 0; t < NIT; ++t) {
        int buf = t & 1;
        const float* wsb = Ws[buf];

        // prefetch node data for next tile (uniform branch, EXEC stays full)
        float4 pj; float vj;
        bool more = (t + 1 < NIT);
        if (more) {
            int j = (t + 1) * 32 + lane;
            pj = pos[j];
            vj = val[j];
        }

        int j0 = t * 32;
#pragma unroll
        for (int kc = 0; kc < 8; ++kc) {
            int kk = kc * 4 + 2 * lhalf;
            float2 a0f = *(const float2*)(&wsb[l15 * WSTR + kk]);
            float2 a1f = *(const float2*)(&wsb[(l15 + 16) * WSTR + kk]);
            const float* bp = targets + (size_t)(j0 + kk) * A + n0 + l15;
            v2f b;  b[0]  = bp[0];  b[1]  = bp[A];
            v2f a0; a0[0] = a0f.x;  a0[1] = a0f.y;
            v2f a1; a1[0] = a1f.x;  a1[1] = a1f.y;
            c0 = __builtin_amdgcn_wmma_f32_16x16x4_f32(false, a0, false, b,
                                                       (short)0, c0, false, false);
            c1 = __builtin_amdgcn_wmma_f32_16x16x4_f32(false, a1, false, b,
                                                       (short)0, c1, false, false);
        }

        if (more) {   // generate tile t+1 into the other buffer
            float dx0 = hx0 - pj.x, dy0 = hy0 - pj.y;
            float dx1 = hx1 - pj.x, dy1 = hy1 - pj.y;
            float w0 = (vi0 * vj) / (sqrtf(dx0 * dx0 + dy0 * dy0) + EPS_DIS);
            float w1 = (vi1 * vj) / (sqrtf(dx1 * dx1 + dy1 * dy1) + EPS_DIS);
            float* wsn = Ws[buf ^ 1];
            wsn[wave * WSTR + lane]        = w0;
            wsn[(wave + 16) * WSTR + lane] = w1;
            rs0 += w0; rs1 += w1;
        }
        __syncthreads();
    }

    // reduce row sums across each wave, publish
#pragma unroll
    for (int off = 16; off > 0; off >>= 1) {
        rs0 += __shfl_xor(rs0, off, 32);
        rs1 += __shfl_xor(rs1, off, 32);
    }
    if (lane == 0) { rs_lds[wave] = rs0; rs_lds[wave + 16] = rs1; }
    __syncthreads();

#pragma unroll
    for (int r = 0; r < 8; ++r) {
        int m = r + 8 * lhalf;
        float inv0 = 1.f / fmaxf(rs_lds[m],      1e-30f);
        float inv1 = 1.f / fmaxf(rs_lds[m + 16], 1e-30f);
        out[(size_t)(m0 + m) * A + n0 + l15]      = c0[r] * inv0;
        out[(size_t)(m0 + 16 + m) * A + n0 + l15] = c1[r] * inv1;
    }
}

// ---------------------------------------------------------------------------
// Host-side launcher
// Inputs (setup_inputs order):
//  0 batch_data [S,F] f32   1 xywh [S,4] f32   2 OW i32   3 OH i32
//  4 actor_weights [S] f32  5 avg_pos [S,2] f32
//  6 W_h [F,2] f32  7 b_h [2] f32  8 W_g [F,A] f32  9 b_g [A] f32
// 10 num_person i32
// Output: targets_weighted [S,A] then hp [S,2], concatenated f32.
// ---------------------------------------------------------------------------
extern "C" void kernel_launch(void* const* d_in, const int* in_sizes, int n_in,
                              void* d_out, int out_size, void* d_ws, size_t ws_size,
                              hipStream_t stream) {
    const float* bd   = (const float*)d_in[0];
    const float* xywh = (const float*)d_in[1];
    const float* Wh   = (const float*)d_in[6];
    const float* bh   = (const float*)d_in[7];
    const float* Wg   = (const float*)d_in[8];
    const float* bg   = (const float*)d_in[9];

    float* out_tw = (float*)d_out;               // [S,A]
    float* out_hp = out_tw + (size_t)S * A;      // [S,2]

    // workspace layout
    float*  ws      = (float*)d_ws;
    float4* pos     = (float4*)ws;               // S float4   (128 KB)
    float*  val     = ws + (size_t)S * 4;        // S floats   ( 32 KB)
    float*  targets = val + S;                   // S*A floats (  8 MB)

    k1_highlight<<<dim3(S / 8), dim3(256), 0, stream>>>(bd, xywh, Wh, bh,
                                                        out_hp, pos, val);
    k2_gcn<<<dim3(S / 16), dim3(512), 0, stream>>>(bd, Wg, bg, targets);
    k3_attn<<<dim3(S / 32), dim3(512), 0, stream>>>(pos, val, targets, out_tw);
}